// Gate_74912819577438
// MI455X (gfx1250) — compile-verified
//
#include <hip/hip_runtime.h>
#include <hip/hip_bf16.h>

// CDNA5 / gfx1250: wave32, WMMA f32 16x16x4.
typedef float v2f __attribute__((ext_vector_type(2)));
typedef float v8f __attribute__((ext_vector_type(8)));

__device__ __forceinline__ v8f wmma4(v2f a, v2f b, v8f c) {
    // 8 args: (neg_a, A, neg_b, B, c_mod, C, reuse_a, reuse_b)
    return __builtin_amdgcn_wmma_f32_16x16x4_f32(
        false, a, false, b, (short)0, c, false, false);
}

// ---------------------------------------------------------------------------
// Build M = (zeta/64) * kron(a + a^T, sigma_x)   (128x128 f32, row major)
// and an identity matrix (used by the fused GEMM for the Taylor/Horner steps).
// H[2i+s, 2j+t] = (s != t) ? (a[i,j] + a[j,i]) : 0
// ---------------------------------------------------------------------------
__global__ void build_kick_h(const float* __restrict__ a_re,
                             const float* __restrict__ zeta,
                             float* __restrict__ M,
                             float* __restrict__ Id) {
    int idx = blockIdx.x * blockDim.x + threadIdx.x;   // 0..16383
    int r = idx >> 7, c = idx & 127;
    int i = r >> 1, s = r & 1;
    int j = c >> 1, t = c & 1;
    float h = (s != t) ? (a_re[i * 64 + j] + a_re[j * 64 + i]) : 0.0f;
    M[idx]  = (zeta[0] * (1.0f / 64.0f)) * h;
    Id[idx] = (r == c) ? 1.0f : 0.0f;
}

// ---------------------------------------------------------------------------
// Fused 128x128 GEMM:  D = alpha * A@B + gamma * I
// One workgroup, 8 waves, each wave owns 8 of the 64 16x16 output tiles.
// f32 WMMA, K stepped by 4. A,B,D are distinct buffers (ping-pong on host).
// ---------------------------------------------------------------------------
__global__ void __launch_bounds__(256)
gemm128(const float* __restrict__ A, const float* __restrict__ B,
        float* __restrict__ D, float alpha, float gamma) {
    const int lane = threadIdx.x & 31;
    const int wave = threadIdx.x >> 5;
    const int nl   = lane & 15;   // N (and M) index within tile
    const int kh   = lane >> 4;   // which K-half of the fragment this lane holds

    for (int t = 0; t < 8; ++t) {
        const int tile = wave * 8 + t;          // 0..63
        const int rowT = (tile >> 3) * 16;
        const int colT = (tile & 7) * 16;
        const int m = rowT + nl;                // A row for this lane
        const int n = colT + nl;                // B col for this lane
        v8f acc = {};
        for (int k0 = 0; k0 < 128; k0 += 4) {
            const int ka = k0 + 2 * kh;
            // A fragment: 16x4, lane holds A[m, ka], A[m, ka+1] (contiguous)
            v2f a = *(const v2f*)(A + m * 128 + ka);
            // B fragment: 4x16, lane holds B[ka, n], B[ka+1, n]
            v2f b;
            b.x = B[ka * 128 + n];
            b.y = B[(ka + 1) * 128 + n];
            acc = wmma4(a, b, acc);
        }
        // C/D layout: VGPR i, lane -> row rowT + 8*kh + i, col n
        for (int i = 0; i < 8; ++i) {
            const int R = rowT + kh * 8 + i;
            float v = alpha * acc[i] + ((R == n) ? gamma : 0.0f);
            D[R * 128 + n] = v;
        }
    }
}

// ---------------------------------------------------------------------------
// Apply: out[b] = Phase(b) ∘ ( (C - iS) @ (xre[b] + i*xim[b]) )
// Batched matvec as GEMM: Y[128 x 1024], tiles 16x16, one tile per wave.
//   Yre = C@xr + S@xi ;  Yim = C@xi - S@xr
// Rotation is a diagonal phase: theta = (X[b]+time) * d[R>>1],
//   d = diag(a@a^dag) = (1..63, 0);  out = (cos - i sin) * Y.
// Output: complex64 interleaved -> one float2 store per element.
// ---------------------------------------------------------------------------
__global__ void __launch_bounds__(256)
apply_kick_rot(const float* __restrict__ C, const float* __restrict__ S,
               const float* __restrict__ xre, const float* __restrict__ xim,
               const float* __restrict__ Xb, const float* __restrict__ tscal,
               float* __restrict__ out) {
    const int lane = threadIdx.x & 31;
    const int wave = threadIdx.x >> 5;
    const int tile = blockIdx.x * 8 + wave;     // 0..511  (8 row-tiles x 64 col-tiles)
    const int rowT = (tile & 7) * 16;
    const int colT = (tile >> 3) * 16;
    const int nl   = lane & 15;
    const int kh   = lane >> 4;
    const int b    = colT + nl;                 // batch index (B-matrix column)
    const int m    = rowT + nl;                 // kick-matrix row (A fragment)

    v8f accR = {};
    v8f accI = {};
    for (int k0 = 0; k0 < 128; k0 += 4) {
        const int ka = k0 + 2 * kh;
        v2f ac = *(const v2f*)(C + m * 128 + ka);
        v2f as = *(const v2f*)(S + m * 128 + ka);
        v2f br = *(const v2f*)(xre + b * 128 + ka);   // states: [B,128] row major
        v2f bi = *(const v2f*)(xim + b * 128 + ka);
        v2f asn; asn.x = -as.x; asn.y = -as.y;        // f32 WMMA NEG only covers C
        accR = wmma4(ac, br, accR);
        accR = wmma4(as, bi, accR);
        accI = wmma4(ac, bi, accI);
        accI = wmma4(asn, br, accI);
    }

    const float dt = Xb[b] + tscal[0];
    float2* o2 = reinterpret_cast<float2*>(out);
    for (int i = 0; i < 8; ++i) {
        const int R = rowT + kh * 8 + i;
        const int f = R >> 1;                         // Fock index
        const float d = (f < 63) ? (float)(f + 1) : 0.0f;
        float sth, cth;
        __sincosf(dt * d, &sth, &cth);
        const float yr = accR[i], yi = accI[i];
        float2 v;
        v.x = cth * yr + sth * yi;                    // (cos - i sin)*(yr + i yi)
        v.y = cth * yi - sth * yr;
        o2[b * 128 + R] = v;
    }
}

// ---------------------------------------------------------------------------
// Host side: expm(-i*zeta*H) by scaling (1/64) + degree-6 Taylor for cos/sin
// + 6 double-angle squaring steps, all with the fused WMMA GEMM. Then one
// fused complex-GEMM + diagonal-phase + store kernel. ~23 tiny launches,
// everything L2 resident, deterministic, graph-capture safe.
// ---------------------------------------------------------------------------
extern "C" void kernel_launch(void* const* d_in, const int* in_sizes, int n_in,
                              void* d_out, int out_size, void* d_ws, size_t ws_size,
                              hipStream_t stream) {
    (void)in_sizes; (void)n_in; (void)out_size; (void)ws_size;
    const float* X    = (const float*)d_in[0];   // [1024]
    const float* sre  = (const float*)d_in[1];   // [1024,128,1]
    const float* sim  = (const float*)d_in[2];   // [1024,128,1]
    const float* are  = (const float*)d_in[3];   // [64,64]
    const float* zeta = (const float*)d_in[4];   // scalar
    const float* tim  = (const float*)d_in[5];   // scalar
    float* out = (float*)d_out;                  // complex64 [1024,128] interleaved

    const size_t SZ = 128 * 128;                 // 64 KB per matrix
    float* w  = (float*)d_ws;
    float* M  = w + 0 * SZ;   // scaled Hamiltonian
    float* Id = w + 1 * SZ;
    float* M2 = w + 2 * SZ;   // M@M
    float* Ca = w + 3 * SZ;   // cos accumulators (ping)
    float* Sa = w + 4 * SZ;   // sin accumulators (ping)
    float* Cb = w + 5 * SZ;   // (pong)
    float* Sb = w + 6 * SZ;   // (pong)
    float* T  = w + 7 * SZ;   // scratch

    build_kick_h<<<64, 256, 0, stream>>>(are, zeta, M, Id);

    // M2 = M@M
    gemm128<<<1, 256, 0, stream>>>(M, M, M2, 1.0f, 0.0f);

    // cos(M) = I - y/2 + y^2/24 - y^3/720   (y = M2), Horner:
    gemm128<<<1, 256, 0, stream>>>(M2, Id, T,  -1.0f / 720.0f, 1.0f / 24.0f);
    gemm128<<<1, 256, 0, stream>>>(T,  M2, Cb,  1.0f,         -0.5f);
    gemm128<<<1, 256, 0, stream>>>(Cb, M2, Ca,  1.0f,          1.0f);

    // sin(M) = M @ (I - y/6 + y^2/120 - y^3/5040), Horner:
    gemm128<<<1, 256, 0, stream>>>(M2, Id, T,  -1.0f / 5040.0f, 1.0f / 120.0f);
    gemm128<<<1, 256, 0, stream>>>(T,  M2, Sb,  1.0f,          -1.0f / 6.0f);
    gemm128<<<1, 256, 0, stream>>>(Sb, M2, T,   1.0f,           1.0f);
    gemm128<<<1, 256, 0, stream>>>(M,  T,  Sa,  1.0f,           0.0f);

    // 6 double-angle steps: cos2 = 2*C@C - I ; sin2 = 2*S@C  (C,S commute)
    for (int s = 0; s < 6; ++s) {
        if ((s & 1) == 0) {
            gemm128<<<1, 256, 0, stream>>>(Sa, Ca, Sb, 2.0f,  0.0f);
            gemm128<<<1, 256, 0, stream>>>(Ca, Ca, Cb, 2.0f, -1.0f);
        } else {
            gemm128<<<1, 256, 0, stream>>>(Sb, Cb, Sa, 2.0f,  0.0f);
            gemm128<<<1, 256, 0, stream>>>(Cb, Cb, Ca, 2.0f, -1.0f);
        }
    }
    // After an even number of steps the final cos/sin are back in Ca / Sa.

    // kick = cos(zeta*H) - i*sin(zeta*H); apply to all 1024 states + phase.
    apply_kick_rot<<<64, 256, 0, stream>>>(Ca, Sa, sre, sim, X, tim, out);
}